// RNN_74629351735828
// MI455X (gfx1250) — compile-verified
//
#include <hip/hip_runtime.h>
#include <math.h>

// Problem sizes (match reference).
#define NH 2048      // hidden
#define NF 512       // features (vocab)
#define NC 512       // classes
#define T  8192      // sequence length

#define THREADS 256
#define WAVES   (THREADS / 32)                 // 8 waves (wave32)

// ---- recurrence decomposition ----
#define RNN_BLOCKS 64
#define ROWS_PER_BLOCK (NH / RNN_BLOCKS)       // 32 rows -> 256KB of W in LDS
#define ROWS_PER_WAVE  (ROWS_PER_BLOCK / WAVES) // 4

// ---- output GEMM tiling ----
#define TM_TILES (T / 16)                      // 512
#define TN_TILES (NC / 16)                     // 32
#define GEMM_BLOCKS (TM_TILES * TN_TILES / WAVES) // 2048

typedef __attribute__((ext_vector_type(2))) float v2f;
typedef __attribute__((ext_vector_type(4))) float v4f;
typedef __attribute__((ext_vector_type(8))) float v8f;
typedef __attribute__((ext_vector_type(4))) unsigned int v4u;
typedef __attribute__((ext_vector_type(4))) int v4i;
typedef __attribute__((ext_vector_type(8))) int v8i;

// ---------------------------------------------------------------------------
// init: reset barrier words, seed h ping-pong buffer with h0.
// (d_ws is poisoned with 0xAA once and never re-poisoned -> self-init.)
// ---------------------------------------------------------------------------
__global__ void rnn_init_kernel(const float* __restrict__ h0,
                                float* __restrict__ hbuf,
                                int* __restrict__ bar) {
  const int i = threadIdx.x;
  if (i == 0) { bar[0] = 0; bar[1] = 0; }
  for (int k = i; k < NH; k += THREADS) hbuf[k] = h0[k];
}

// ---------------------------------------------------------------------------
// device-wide sense barrier (64 co-resident blocks, acq/rel at agent scope)
// ---------------------------------------------------------------------------
__device__ __forceinline__ void gridBarrier(int* cnt, int* gen, int nblocks,
                                            int tid) {
  __syncthreads();
  if (tid == 0) {
    __threadfence();  // make this block's h stores device-visible
    const int g = __hip_atomic_load(gen, __ATOMIC_RELAXED,
                                    __HIP_MEMORY_SCOPE_AGENT);
    const int arrived = __hip_atomic_fetch_add(cnt, 1, __ATOMIC_ACQ_REL,
                                               __HIP_MEMORY_SCOPE_AGENT);
    if (arrived == nblocks - 1) {
      __hip_atomic_store(cnt, 0, __ATOMIC_RELAXED, __HIP_MEMORY_SCOPE_AGENT);
      __hip_atomic_fetch_add(gen, 1, __ATOMIC_RELEASE,
                             __HIP_MEMORY_SCOPE_AGENT);
    } else {
      while (__hip_atomic_load(gen, __ATOMIC_ACQUIRE,
                               __HIP_MEMORY_SCOPE_AGENT) == g) {
        __builtin_amdgcn_s_sleep(1);
      }
    }
  }
  __syncthreads();
}

// ---------------------------------------------------------------------------
// Persistent recurrence kernel: h_{t+1} = tanh(W h_t + U[:, id_t]).
// Each block caches its 32 rows of W in LDS once (256KB of the 320KB/WGP) via
// the Tensor Data Mover, then per step only h (8KB) moves through L2 + one
// device barrier.
// ---------------------------------------------------------------------------
__global__ void __launch_bounds__(THREADS, 1)
rnn_recurrence_kernel(const int* __restrict__ ids,
                      const float* __restrict__ W,
                      const float* __restrict__ U,
                      float* hbuf,            // [2][NH] ping-pong
                      float* __restrict__ Hout,   // [T][NH] all hidden states
                      float* __restrict__ h_final,// d_out[0:NH]
                      int* bar) {
  __shared__ float w_lds[ROWS_PER_BLOCK * NH];  // 256 KB
  __shared__ float h_lds[NH];                   // 8 KB

  const int tid = threadIdx.x;
  const int wave = tid >> 5;
  const int lane = tid & 31;
  const int rowBase = blockIdx.x * ROWS_PER_BLOCK;
  const int rloc0 = wave * ROWS_PER_WAVE;

#if __has_builtin(__builtin_amdgcn_tensor_load_to_lds)
  // One-time TDM preload: 2D tile (tile_dim0=2048 elems, tile_dim1=32 rows),
  // 4-byte elements, row stride 2048, global -> LDS. D# per ISA 8.3/8.4.
  if (wave == 0) {
    const unsigned long long gaddr =
        (unsigned long long)(uintptr_t)(W + (size_t)rowBase * NH);
    const unsigned lds_off = (unsigned)(uintptr_t)(void*)&w_lds[0];
    v4u g0;
    g0[0] = 1u;                                            // count=1, flags=0
    g0[1] = lds_off;                                       // lds_addr
    g0[2] = (unsigned)(gaddr & 0xFFFFFFFFu);               // global_addr lo
    g0[3] = (unsigned)((gaddr >> 32) & 0x01FFFFFFu) | (2u << 30);  // hi|type=2
    v8i g1;
    g1[0] = (int)(2u << 16);          // workgroup_mask=0, data_size=2 (4B)
    g1[1] = (int)(2048u << 16);       // tensor_dim0[15:0]=2048
    g1[2] = (int)(32u << 16);         // tensor_dim0 hi=0 | tensor_dim1=32
    g1[3] = (int)(2048u << 16);       // tensor_dim1 hi=0 | tile_dim0=2048
    g1[4] = 32;                       // tile_dim1=32, tile_dim2=0
    g1[5] = 2048;                     // tensor_dim0_stride lo32 = 2048
    g1[6] = 0;                        // stride hi16=0, tensor_dim1_stride=0
    g1[7] = 0;
    v4i z4 = {0, 0, 0, 0};
#if __clang_major__ >= 23
    v8i z8 = {0, 0, 0, 0, 0, 0, 0, 0};
    __builtin_amdgcn_tensor_load_to_lds(g0, g1, z4, z4, z8, 0);
#else
    __builtin_amdgcn_tensor_load_to_lds(g0, g1, z4, z4, 0);
#endif
    __builtin_amdgcn_s_wait_tensorcnt(0);
  }
#else
  // Fallback: coalesced b128 copy loop.
  {
    const float4* src = (const float4*)(W + (size_t)rowBase * NH);
    float4* dst = (float4*)w_lds;
    const int n4 = ROWS_PER_BLOCK * NH / 4;
    for (int i = tid; i < n4; i += THREADS) dst[i] = src[i];
  }
#endif
  __syncthreads();

  for (int t = 0; t < T; ++t) {
    const float* hprev = hbuf + (size_t)(t & 1) * NH;
    float* hnext = hbuf + (size_t)((t + 1) & 1) * NH;

    // Pull h_t into LDS; agent-scope loads force scope=DEV (skip stale WGP$).
    for (int i = tid; i < NH; i += THREADS)
      h_lds[i] = __hip_atomic_load(&hprev[i], __ATOMIC_RELAXED,
                                   __HIP_MEMORY_SCOPE_AGENT);
    __syncthreads();

    // 4 rows per wave, lane-strided dot product out of LDS.
    float acc[ROWS_PER_WAVE] = {0.f, 0.f, 0.f, 0.f};
    for (int k = lane; k < NH; k += 32) {
      const float hv = h_lds[k];
#pragma unroll
      for (int rr = 0; rr < ROWS_PER_WAVE; ++rr)
        acc[rr] = fmaf(w_lds[(rloc0 + rr) * NH + k], hv, acc[rr]);
    }

    const int id_t = ids[t];
#pragma unroll
    for (int rr = 0; rr < ROWS_PER_WAVE; ++rr) {
      float v = acc[rr];
#pragma unroll
      for (int s = 16; s > 0; s >>= 1) v += __shfl_xor(v, s, 32);
      if (lane == 0) {
        const int r = rowBase + rloc0 + rr;
        const float hv = tanhf(v + U[(size_t)r * NF + id_t]);
        hnext[r] = hv;                    // next step's input
        Hout[(size_t)t * NH + r] = hv;    // saved for output GEMM
        if (t == T - 1) h_final[r] = hv;  // reference output #1
      }
    }

    gridBarrier(bar, bar + 1, RNN_BLOCKS, tid);
  }
}

// ---------------------------------------------------------------------------
// O[T][NC] = H[T][NH] x V[NC][NH]^T via V_WMMA_F32_16X16X4_F32 (fp32 WMMA).
// One 16x16 C tile per wave; K loop: one b128 per matrix feeds two WMMAs
// (K-permutation is legal since the dot product commutes over k, and A/B use
// the identical permutation).
// VGPR layouts per CDNA5 ISA 7.12.2 (g = lane/16, l = lane%16):
//   A(16x4):  a[j] = A(row=l, k = perm(kk, g, j))
//   B(4x16):  b[j] = B(k = perm(kk, g, j), col=l)
//   C/D:      acc[v] -> row tm*16 + v + 8g, col tn*16 + l
// ---------------------------------------------------------------------------
__global__ void __launch_bounds__(THREADS)
rnn_outproj_wmma_kernel(const float* __restrict__ H,
                        const float* __restrict__ V,
                        float* __restrict__ O) {
  const int gwave = blockIdx.x * WAVES + (threadIdx.x >> 5);
  const int lane = threadIdx.x & 31;
  const int tm = gwave / TN_TILES;
  const int tn = gwave % TN_TILES;
  const int g = lane >> 4;
  const int l = lane & 15;

  const float* arow = H + (size_t)(tm * 16 + l) * NH + 4 * g;  // time rows
  const float* brow = V + (size_t)(tn * 16 + l) * NH + 4 * g;  // class rows

  v8f acc = {};
  for (int kk = 0; kk < NH; kk += 8) {
    v4f a = *(const v4f*)(arow + kk);   // one b128 per matrix per 2 WMMAs
    v4f b = *(const v4f*)(brow + kk);
    v2f a0 = {a[0], a[1]}, a1 = {a[2], a[3]};
    v2f b0 = {b[0], b[1]}, b1 = {b[2], b[3]};
    acc = __builtin_amdgcn_wmma_f32_16x16x4_f32(
        false, a0, false, b0, (short)0, acc, false, false);
    acc = __builtin_amdgcn_wmma_f32_16x16x4_f32(
        false, a1, false, b1, (short)0, acc, false, false);
  }

  const int col = tn * 16 + l;
#pragma unroll
  for (int v = 0; v < 8; ++v) {
    const int row = tm * 16 + v + 8 * g;
    O[(size_t)row * NC + col] = acc[v];
  }
}

// ---------------------------------------------------------------------------
extern "C" void kernel_launch(void* const* d_in, const int* in_sizes, int n_in,
                              void* d_out, int out_size, void* d_ws,
                              size_t ws_size, hipStream_t stream) {
  const float* h0  = (const float*)d_in[0];  // [NH]
  const int*   ids = (const int*)d_in[1];    // [T]
  const float* W   = (const float*)d_in[2];  // [NH][NH]
  const float* U   = (const float*)d_in[3];  // [NH][NF]
  const float* V   = (const float*)d_in[4];  // [NC][NH]
  float* out = (float*)d_out;                // [NH] h_final, then [T][NC]

  // Workspace layout: H (64MB) | h ping-pong (16KB) | barrier words.
  float* Hbuf = (float*)d_ws;                // T*NH floats
  float* hbuf = Hbuf + (size_t)T * NH;       // 2*NH floats
  int*   bar  = (int*)(hbuf + 2 * NH);       // 2 ints

  rnn_init_kernel<<<1, THREADS, 0, stream>>>(h0, hbuf, bar);
  rnn_recurrence_kernel<<<RNN_BLOCKS, THREADS, 0, stream>>>(
      ids, W, U, hbuf, Hbuf, out, bar);
  rnn_outproj_wmma_kernel<<<GEMM_BLOCKS, THREADS, 0, stream>>>(
      Hbuf, V, out + NH);
  (void)in_sizes; (void)n_in; (void)out_size; (void)ws_size;
}